// Encoder_24111946400020
// MI455X (gfx1250) — compile-verified
//
#include <hip/hip_runtime.h>

#define N_NODES 50000
#define N_EDGES 600000
#define D_IN    5
#define D_H     128
#define BN_EPS  1e-5f
#define SCAN_B  256

typedef __attribute__((ext_vector_type(2))) float v2f;
typedef __attribute__((ext_vector_type(8))) float v8f;

// ---------------- utility ----------------
__global__ void zero_kernel(float* __restrict__ p, int n) {
    int i = blockIdx.x * blockDim.x + threadIdx.x;
    if (i < n) p[i] = 0.0f;
}

// ---------------- CSR construction (dst-indexed), built once ----------------
__global__ void count_kernel(const int* __restrict__ dst, int* __restrict__ cnt, int E) {
    int e = blockIdx.x * blockDim.x + threadIdx.x;
    if (e < E) atomicAdd(&cnt[dst[e]], 1);
}

__global__ void blocksum_kernel(const int* __restrict__ cnt, int* __restrict__ bsum, int n) {
    __shared__ int sh[SCAN_B];
    int t = threadIdx.x;
    int i = blockIdx.x * SCAN_B + t;
    sh[t] = (i < n) ? cnt[i] : 0;
    __syncthreads();
    for (int off = SCAN_B / 2; off > 0; off >>= 1) {
        if (t < off) sh[t] += sh[t + off];
        __syncthreads();
    }
    if (t == 0) bsum[blockIdx.x] = sh[0];
}

// single block: exclusive scan of bsum[nb] in place (nb <= 1024)
__global__ void scan_bsum_kernel(int* __restrict__ bsum, int nb) {
    __shared__ int sh[1024];
    int t = threadIdx.x;
    int v = (t < nb) ? bsum[t] : 0;
    sh[t] = v;
    __syncthreads();
    for (int off = 1; off < 1024; off <<= 1) {
        int a = (t >= off) ? sh[t - off] : 0;
        __syncthreads();
        sh[t] += a;
        __syncthreads();
    }
    if (t < nb) bsum[t] = sh[t] - v;   // exclusive
}

__global__ void blockscan_kernel(const int* __restrict__ cnt, const int* __restrict__ bsum,
                                 int* __restrict__ row_start, int n) {
    __shared__ int sh[SCAN_B];
    int t = threadIdx.x;
    int i = blockIdx.x * SCAN_B + t;
    int v = (i < n) ? cnt[i] : 0;
    sh[t] = v;
    __syncthreads();
    for (int off = 1; off < SCAN_B; off <<= 1) {
        int a = (t >= off) ? sh[t - off] : 0;
        __syncthreads();
        sh[t] += a;
        __syncthreads();
    }
    if (i < n) row_start[i] = bsum[blockIdx.x] + sh[t] - v;     // exclusive
    if (i == n - 1) row_start[n] = bsum[blockIdx.x] + sh[t];    // == E
}

__global__ void fill_csr_kernel(const int* __restrict__ dst, const int* __restrict__ row_start,
                                int* __restrict__ cursor, int* __restrict__ csr, int E) {
    int e = blockIdx.x * blockDim.x + threadIdx.x;
    if (e < E) {
        int d = dst[e];
        int p = atomicAdd(&cursor[d], 1);
        csr[row_start[d] + p] = e;
    }
}

// dis[n] = rsqrt(1 + sum_{e: dst -> n} w[e])   (atomic-free via CSR)
__global__ void dis_kernel(const int* __restrict__ row_start, const int* __restrict__ csr,
                           const float* __restrict__ w, float* __restrict__ dis, int n) {
    int i = blockIdx.x * blockDim.x + threadIdx.x;
    if (i < n) {
        float d = 1.0f;   // self-loop weight
        int beg = row_start[i], end = row_start[i + 1];
        for (int e = beg; e < end; ++e) d += w[csr[e]];
        dis[i] = rsqrtf(d);
    }
}

// ---------------- BatchNorm helpers ----------------
// generic per-column stats (used for BN0, C=5): blockDim.x == C
__global__ void colstats_kernel(const float* __restrict__ x, int n, int C,
                                float* __restrict__ sum, float* __restrict__ sqsum) {
    int c = threadIdx.x;
    float s = 0.f, q = 0.f;
    for (int r = blockIdx.x; r < n; r += gridDim.x) {
        float v = x[(size_t)r * C + c];
        s += v;
        q += v * v;
    }
    atomicAdd(&sum[c], s);
    atomicAdd(&sqsum[c], q);
}

// 128-channel stats: single wave per block, 4 channels per lane, b128 row reads
__global__ void colstats128_kernel(const float* __restrict__ x, int n,
                                   float* __restrict__ sum, float* __restrict__ sqsum) {
    int lane = threadIdx.x;        // 0..31
    int c4 = lane << 2;
    float sx = 0.f, sy = 0.f, sz = 0.f, sw = 0.f;
    float qx = 0.f, qy = 0.f, qz = 0.f, qw = 0.f;
    for (int r = blockIdx.x; r < n; r += gridDim.x) {
        float4 v = *(const float4*)(x + (size_t)r * D_H + c4);
        sx += v.x; sy += v.y; sz += v.z; sw += v.w;
        qx += v.x * v.x; qy += v.y * v.y; qz += v.z * v.z; qw += v.w * v.w;
    }
    atomicAdd(&sum[c4 + 0], sx);  atomicAdd(&sum[c4 + 1], sy);
    atomicAdd(&sum[c4 + 2], sz);  atomicAdd(&sum[c4 + 3], sw);
    atomicAdd(&sqsum[c4 + 0], qx); atomicAdd(&sqsum[c4 + 1], qy);
    atomicAdd(&sqsum[c4 + 2], qz); atomicAdd(&sqsum[c4 + 3], qw);
}

// scale = g * rsqrt(var + eps); shift = beta - mean*scale   (biased variance)
__global__ void finalize_bn_kernel(const float* __restrict__ sum, const float* __restrict__ sqsum,
                                   const float* __restrict__ g, const float* __restrict__ beta,
                                   int C, float n, float* __restrict__ scale,
                                   float* __restrict__ shift) {
    int c = threadIdx.x;
    if (c < C) {
        float m = sum[c] / n;
        float v = sqsum[c] / n - m * m;
        float s = g[c] * rsqrtf(v + BN_EPS);
        scale[c] = s;
        shift[c] = beta[c] - m * s;
    }
}

// out[n][c] = sum_k (h[n][k]*scale0[k]+shift0[k]) * W1[k][c]
// single wave per node, 4 channels per lane (b128 loads of W1 rows, b128 store)
__global__ void bn0_lin1_kernel(const float* __restrict__ h, const float* __restrict__ scale0,
                                const float* __restrict__ shift0, const float* __restrict__ W1,
                                float* __restrict__ out) {
    int n = blockIdx.x;
    int lane = threadIdx.x;        // 0..31
    int c4 = lane << 2;
    float hk = 0.f;
    if (lane < D_IN) hk = h[(size_t)n * D_IN + lane] * scale0[lane] + shift0[lane];
    float ax = 0.f, ay = 0.f, az = 0.f, aw = 0.f;
#pragma unroll
    for (int k = 0; k < D_IN; ++k) {
        float hv = __shfl(hk, k, 32);                       // broadcast BN0(h[n][k])
        float4 wv = *(const float4*)(W1 + k * D_H + c4);
        ax += hv * wv.x; ay += hv * wv.y; az += hv * wv.z; aw += hv * wv.w;
    }
    float4 o; o.x = ax; o.y = ay; o.z = az; o.w = aw;
    *(float4*)(out + (size_t)n * D_H + c4) = o;
}

// GCN aggregation as a pure gather: single wave per node, 4 channels per lane.
// conv[n][c] = dis[n]^2 * xw[n][c] + sum_{e in CSR[n]} dis[src]*w*dis[n] * xw[src][c]
// Edge (src, norm) pairs live in registers; broadcast via __shfl (no LDS, no barriers).
__global__ void gather_conv_kernel(const float* __restrict__ xw, const float* __restrict__ dis,
                                   const int* __restrict__ src, const float* __restrict__ w,
                                   const int* __restrict__ row_start, const int* __restrict__ csr,
                                   float* __restrict__ conv) {
    int n = blockIdx.x;
    int lane = threadIdx.x;        // 0..31
    int c4 = lane << 2;
    float dn = dis[n];
    float4 sv = *(const float4*)(xw + (size_t)n * D_H + c4);
    float d2 = dn * dn;
    float ax = d2 * sv.x, ay = d2 * sv.y, az = d2 * sv.z, aw = d2 * sv.w;  // self-loop
    int beg = row_start[n], end = row_start[n + 1];
    for (int chunk = beg; chunk < end; chunk += 32) {
        int m = min(end - chunk, 32);
        int   s_r = 0;
        float nrm_r = 0.f;
        if (lane < m) {
            int e = csr[chunk + lane];
            s_r = src[e];
            nrm_r = dis[s_r] * w[e] * dn;
        }
        for (int j = 0; j < m; ++j) {
            int   s  = __shfl(s_r, j, 32);
            float nm = __shfl(nrm_r, j, 32);
            float4 r = *(const float4*)(xw + (size_t)s * D_H + c4);   // b128, coalesced
            ax += nm * r.x; ay += nm * r.y; az += nm * r.z; aw += nm * r.w;
        }
    }
    float4 o; o.x = ax; o.y = ay; o.z = az; o.w = aw;
    *(float4*)(conv + (size_t)n * D_H + c4) = o;
}

// out = relu(conv*scale[c] + shift[c]), float4-vectorized
__global__ void bnrelu_kernel(const float* __restrict__ conv, const float* __restrict__ scale,
                              const float* __restrict__ shift, float* __restrict__ out, int total4) {
    int i = blockIdx.x * blockDim.x + threadIdx.x;
    if (i < total4) {
        int c4 = (i << 2) & (D_H - 1);
        float4 v = ((const float4*)conv)[i];
        float4 s = *(const float4*)(scale + c4);
        float4 t = *(const float4*)(shift + c4);
        float4 o;
        o.x = fmaxf(v.x * s.x + t.x, 0.f);
        o.y = fmaxf(v.y * s.y + t.y, 0.f);
        o.z = fmaxf(v.z * s.z + t.z, 0.f);
        o.w = fmaxf(v.w * s.w + t.w, 0.f);
        ((float4*)out)[i] = o;
    }
}

// out[M,128] = A[M,128] @ W[128,128] (+bias), full-fp32 WMMA.
// One wave32 per 16x16 output tile; K-loop of 32 x V_WMMA_F32_16X16X4_F32.
__global__ void wmma_gemm_kernel(const float* __restrict__ A, const float* __restrict__ W,
                                 const float* __restrict__ bias, float* __restrict__ out) {
    int lane   = threadIdx.x;          // 0..31, EXEC all ones
    int m_base = blockIdx.y * 16;
    int n_base = blockIdx.x * 16;
    int mr = lane & 15;                // row (A) / col (B,C,D) within tile
    int kh = (lane >> 4) << 1;         // 0 or 2: which K pair this half-wave holds
    const float* arow = A + (size_t)(m_base + mr) * D_H + kh;
    v8f acc = {};
#pragma unroll
    for (int kk = 0; kk < D_H; kk += 4) {
        v2f a = *(const v2f*)(arow + kk);               // A[m][kk+kh], A[m][kk+kh+1]
        v2f b;
        b.x = W[(kk + kh + 0) * D_H + n_base + mr];     // B[kk+kh  ][n]
        b.y = W[(kk + kh + 1) * D_H + n_base + mr];     // B[kk+kh+1][n]
        acc = __builtin_amdgcn_wmma_f32_16x16x4_f32(
            /*neg_a=*/false, a, /*neg_b=*/false, b,
            /*c_mod=*/(short)0, acc, /*reuse_a=*/false, /*reuse_b=*/false);
    }
    int col  = n_base + mr;
    int row0 = m_base + ((lane >> 4) << 3);            // 0..7 (lo half) / 8..15 (hi half)
    float bb = bias ? bias[col] : 0.f;
#pragma unroll
    for (int i = 0; i < 8; ++i)
        out[(size_t)(row0 + i) * D_H + col] = acc[i] + bb;
}

extern "C" void kernel_launch(void* const* d_in, const int* in_sizes, int n_in,
                              void* d_out, int out_size, void* d_ws, size_t ws_size,
                              hipStream_t stream) {
    (void)in_sizes; (void)n_in; (void)out_size; (void)ws_size;

    const float* h    = (const float*)d_in[0];
    const int*   eidx = (const int*)d_in[1];
    const float* ew   = (const float*)d_in[2];
    const float* g0   = (const float*)d_in[3];
    const float* be0  = (const float*)d_in[4];
    const float* W1   = (const float*)d_in[5];
    /* b1 = d_in[6]: bias followed by training-mode BN is a no-op -> dropped */
    const float* g1   = (const float*)d_in[7];
    const float* be1  = (const float*)d_in[8];
    const float* W2   = (const float*)d_in[9];
    /* b2 = d_in[10]: dropped (BN) */
    const float* g2   = (const float*)d_in[11];
    const float* be2  = (const float*)d_in[12];
    const float* Wmu  = (const float*)d_in[13];
    const float* bmu  = (const float*)d_in[14];
    const float* Wls  = (const float*)d_in[15];
    const float* bls  = (const float*)d_in[16];

    const int* src = eidx;              // edge_index[0]
    const int* dst = eidx + N_EDGES;    // edge_index[1]

    float* ws = (float*)d_ws;
    size_t NH = (size_t)N_NODES * D_H;
    float* A    = ws;             // xW   (linear-transform output)
    float* B    = ws + NH;        // conv (aggregation output)
    float* C    = ws + 2 * NH;    // h    (post BN+ReLU features)
    float* dis  = ws + 3 * NH;    // [N]
    float* ssum = dis + N_NODES;  // [128]
    float* ssq  = ssum + D_H;     // [128]
    float* scl  = ssq + D_H;      // [128]
    float* sft  = scl + D_H;      // [128]
    int*   cnt       = (int*)(sft + D_H);          // [N]
    int*   row_start = cnt + N_NODES;              // [N+1]
    int*   cursor    = row_start + N_NODES + 1;    // [N]
    int*   bsum      = cursor + N_NODES;           // [<=1024]
    int*   csr       = bsum + 1024;                // [E]

    float* mu_out = (float*)d_out;
    float* ls_out = mu_out + NH;

    int nscan = (N_NODES + SCAN_B - 1) / SCAN_B;   // 196 blocks

    // ---- build dst-indexed CSR once (shared by both conv layers) ----
    zero_kernel<<<(2 * N_NODES + 255) / 256, 256, 0, stream>>>((float*)cnt, 2 * N_NODES);
    zero_kernel<<<(N_NODES + 255) / 256, 256, 0, stream>>>((float*)cursor, N_NODES);
    count_kernel<<<(N_EDGES + 255) / 256, 256, 0, stream>>>(dst, cnt, N_EDGES);
    blocksum_kernel<<<nscan, SCAN_B, 0, stream>>>(cnt, bsum, N_NODES);
    scan_bsum_kernel<<<1, 1024, 0, stream>>>(bsum, nscan);
    blockscan_kernel<<<nscan, SCAN_B, 0, stream>>>(cnt, bsum, row_start, N_NODES);
    fill_csr_kernel<<<(N_EDGES + 255) / 256, 256, 0, stream>>>(dst, row_start, cursor, csr, N_EDGES);

    // ---- symmetric normalization coefficients ----
    dis_kernel<<<(N_NODES + 255) / 256, 256, 0, stream>>>(row_start, csr, ew, dis, N_NODES);

    // ---- BN0 (5 channels) ----
    zero_kernel<<<1, 2 * D_H, 0, stream>>>(ssum, 2 * D_H);
    colstats_kernel<<<512, D_IN, 0, stream>>>(h, N_NODES, D_IN, ssum, ssq);
    finalize_bn_kernel<<<1, D_H, 0, stream>>>(ssum, ssq, g0, be0, D_IN, (float)N_NODES, scl, sft);

    // ---- layer 1: fused BN0 + (5->128) linear, CSR-gather aggregation ----
    bn0_lin1_kernel<<<N_NODES, 32, 0, stream>>>(h, scl, sft, W1, A);
    gather_conv_kernel<<<N_NODES, 32, 0, stream>>>(A, dis, src, ew, row_start, csr, B);
    zero_kernel<<<1, 2 * D_H, 0, stream>>>(ssum, 2 * D_H);
    colstats128_kernel<<<512, 32, 0, stream>>>(B, N_NODES, ssum, ssq);
    finalize_bn_kernel<<<1, D_H, 0, stream>>>(ssum, ssq, g1, be1, D_H, (float)N_NODES, scl, sft);
    bnrelu_kernel<<<(int)((NH / 4 + 255) / 256), 256, 0, stream>>>(B, scl, sft, C, (int)(NH / 4));

    // ---- layer 2: WMMA GEMM + CSR-gather aggregation ----
    dim3 ggrid(D_H / 16, N_NODES / 16);   // 8 x 3125 wave-tiles
    wmma_gemm_kernel<<<ggrid, 32, 0, stream>>>(C, W2, nullptr, A);
    gather_conv_kernel<<<N_NODES, 32, 0, stream>>>(A, dis, src, ew, row_start, csr, B);
    zero_kernel<<<1, 2 * D_H, 0, stream>>>(ssum, 2 * D_H);
    colstats128_kernel<<<512, 32, 0, stream>>>(B, N_NODES, ssum, ssq);
    finalize_bn_kernel<<<1, D_H, 0, stream>>>(ssum, ssq, g2, be2, D_H, (float)N_NODES, scl, sft);
    bnrelu_kernel<<<(int)((NH / 4 + 255) / 256), 256, 0, stream>>>(B, scl, sft, C, (int)(NH / 4));

    // ---- heads: mu and log_std straight into d_out ----
    wmma_gemm_kernel<<<ggrid, 32, 0, stream>>>(C, Wmu, bmu, mu_out);
    wmma_gemm_kernel<<<ggrid, 32, 0, stream>>>(C, Wls, bls, ls_out);
}